// MultiScaleSTGCN_38113539785277
// MI455X (gfx1250) — compile-verified
//
#include <hip/hip_runtime.h>

typedef __attribute__((ext_vector_type(2))) float v2f;
typedef __attribute__((ext_vector_type(8))) float v8f;
typedef __attribute__((ext_vector_type(4))) unsigned int u32x4;
typedef __attribute__((ext_vector_type(8))) int i32x8;
typedef __attribute__((ext_vector_type(4))) int i32x4;

#define BDIM 32
#define SDIM 256
#define NDIM 128
#define CIN  32
#define HDIM 64
#define HB   16
#define EPSF 1e-5f

// padded LDS pitches (floats): keep 16B alignment, spread WMMA A-fragment
// lanes across banks (pitch mod 64 = 4 or 36 -> 16 distinct banks)
#define AN_LD  132   // 128 + 4
#define X_LD   36    // 32 + 4
#define SEQ_LD 68    // 64 + 4

#if defined(__HIP_DEVICE_COMPILE__) && __has_builtin(__builtin_amdgcn_tensor_load_to_lds) && __has_builtin(__builtin_amdgcn_s_wait_tensorcnt)
#define USE_TDM 1
#else
#define USE_TDM 0
#endif

#if USE_TDM
// ---------------------------------------------------------------------------
// TDM: DMA a 2D strided tile (rows x dim0 elements, f32) from global into LDS
// with hardware row padding (pad_interval/pad_amount reproduce our padded
// pitches). D# packing per CDNA5 ISA 8.3/8.4: count=1, type=2, data_size=4B,
// 2D tile -> groups 2/3 zero. Issue from one lane; EXEC is ignored by TDM.
// This toolchain exposes the 6-arg builtin (g0, g1, g2, g3, g4, cpol).
// ---------------------------------------------------------------------------
__device__ __forceinline__ void tdm_load_2d(unsigned lds_byte_addr,
                                            const void* gptr,
                                            unsigned dim0, unsigned rows,
                                            unsigned long long stride0,
                                            unsigned pad_interval,
                                            unsigned pad_amount) {
  unsigned long long ga = (unsigned long long)(uintptr_t)gptr;
  u32x4 g0;
  g0[0] = 1u;                                              // count=1, user mode
  g0[1] = lds_byte_addr;                                   // lds_addr
  g0[2] = (unsigned)(ga & 0xFFFFFFFFu);                    // global_addr[31:0]
  g0[3] = (unsigned)((ga >> 32) & 0x01FFFFFFu) | (2u << 30);  // addr[56:32]|type=2
  i32x8 g1;
  g1[0] = (int)((2u << 16)                                 // data_size = 4B
              | (1u << 20)                                 // pad_enable
              | (pad_interval << 22) | (pad_amount << 25));
  g1[1] = (int)((dim0 & 0xFFFFu) << 16);                   // tensor_dim0[15:0]
  g1[2] = (int)((dim0 >> 16) | ((rows & 0xFFFFu) << 16));  // dim0[31:16]|dim1[15:0]
  g1[3] = (int)((rows >> 16) | ((dim0 & 0xFFFFu) << 16));  // dim1[31:16]|tile_dim0
  g1[4] = (int)(rows & 0xFFFFu);                           // tile_dim1 (tile_dim2=0)
  g1[5] = (int)(stride0 & 0xFFFFFFFFull);                  // dim0_stride[31:0]
  g1[6] = (int)((stride0 >> 32) & 0xFFFFull);              // dim0_stride[47:32]
  g1[7] = 0;                                               // dim1_stride unused (2D)
  i32x4 g2 = {0, 0, 0, 0}, g3 = {0, 0, 0, 0};
  i32x8 g4 = {0, 0, 0, 0, 0, 0, 0, 0};
  __builtin_amdgcn_tensor_load_to_lds(g0, g1, g2, g3, g4, 0);
}
#endif

// ---------------------------------------------------------------------------
// Kernel 1: An = D^-1/2 (A + I) D^-1/2   (128x128, one block, L2-resident)
// ---------------------------------------------------------------------------
__global__ void stgcn_adj_norm(const float* __restrict__ adj, float* __restrict__ An) {
  __shared__ float dis[NDIM];
  const int m = threadIdx.x;           // 128 threads
  float s = 1.0f;                      // +I on the diagonal
  for (int nn = 0; nn < NDIM; ++nn) s += adj[m * NDIM + nn];
  dis[m] = (s > 0.f) ? rsqrtf(s) : 0.f;
  __syncthreads();
  const float dm = dis[m];
  for (int nn = 0; nn < NDIM; ++nn) {
    float a = adj[m * NDIM + nn] + ((nn == m) ? 1.f : 0.f);
    An[m * NDIM + nn] = dm * a * dis[nn];
  }
}

// ---------------------------------------------------------------------------
// Kernel 2: per (b,s): ax = An@x, atx = An^T@x (WMMA, K=128), then
// xs = [x|ax|atx] @ [theta;phi;psi] + bias     (WMMA, K=96, N=64)
// 512 threads = 16 waves; ~148KB LDS -> 2 workgroups per WGP.
// ---------------------------------------------------------------------------
__global__ void __launch_bounds__(512) stgcn_spatial(
    const float* __restrict__ x, const float* __restrict__ An,
    const float* __restrict__ thw, const float* __restrict__ thb,
    const float* __restrict__ phw, const float* __restrict__ phb,
    const float* __restrict__ psw, const float* __restrict__ psb,
    float* __restrict__ xs)
{
  extern __shared__ float smem[];
  float* sAn  = smem;                       // 128*132
  float* sx   = sAn  + NDIM * AN_LD;        // 128*36
  float* sax  = sx   + NDIM * X_LD;         // 128*36
  float* satx = sax  + NDIM * X_LD;         // 128*36
  float* sW   = satx + NDIM * X_LD;         // 96*64 fused [theta;phi;psi]
  float* sB   = sW   + 3 * CIN * HDIM;      // 64 fused bias

  const int tid   = threadIdx.x;            // 512 = 16 waves
  const int wave  = tid >> 5;
  const int lane  = tid & 31;
  const int row16 = lane & 15;
  const int khalf = (lane >> 4) * 2;        // K sub-offset per lane half
  const int col   = lane & 15;

  // --- stage An (padded) and x tile (padded): TDM DMA with HW padding ---
#if USE_TDM
  if (tid == 0) {
    tdm_load_2d((unsigned)(uintptr_t)sAn, An, NDIM, NDIM,
                (unsigned long long)NDIM, /*pi=128dw*/6, /*pa=4dw*/3);
    tdm_load_2d((unsigned)(uintptr_t)sx,
                x + (size_t)blockIdx.x * NDIM * CIN, CIN, NDIM,
                (unsigned long long)CIN, /*pi=32dw*/4, /*pa=4dw*/3);
  }
#else
  for (int i = tid; i < NDIM * NDIM / 4; i += 512) {      // An: 128 rows x 32 quads
    const int r = i >> 5, q = (i & 31) * 4;
    *(float4*)(sAn + r * AN_LD + q) = *(const float4*)(An + r * NDIM + q);
  }
  {
    const float* xsrc = x + (size_t)blockIdx.x * NDIM * CIN;
    for (int i = tid; i < NDIM * CIN / 4; i += 512) {     // x: 128 rows x 8 quads
      const int r = i >> 3, q = (i & 7) * 4;
      *(float4*)(sx + r * X_LD + q) = *(const float4*)(xsrc + r * CIN + q);
    }
  }
#endif
  for (int i = tid; i < 3 * CIN * HDIM / 4; i += 512) {   // W: 96 rows x 16 quads
    const int r = i >> 4, q = (i & 15) * 4;
    const float* src = (r < CIN) ? (thw + r * HDIM)
                     : (r < 2 * CIN) ? (phw + (r - CIN) * HDIM)
                                     : (psw + (r - 2 * CIN) * HDIM);
    *(float4*)(sW + r * HDIM + q) = *(const float4*)(src + q);
  }
  if (tid < HDIM) sB[tid] = thb[tid] + phb[tid] + psb[tid];
#if USE_TDM
  __builtin_amdgcn_s_wait_tensorcnt(0);
#endif
  __syncthreads();

  // --- GEMM1: wave w -> mat = w>>3 (uniform), tileM = w&7, both tileN ---
  {
    const int mat   = wave >> 3;
    const int tileM = wave & 7;
    const int row   = tileM * 16 + row16;
    const float* b0p = sx + khalf * X_LD + col;        // tileN = 0
    const float* b1p = b0p + 16;                       // tileN = 1
    v8f acc0 = {}, acc1 = {};
    if (mat == 0) {                                    // ax = An @ x
      const float* ap = sAn + row * AN_LD + khalf;
      #pragma unroll
      for (int k0 = 0; k0 < NDIM; k0 += 4) {
        v2f a = *(const v2f*)(ap + k0);
        v2f b0; b0.x = b0p[k0 * X_LD]; b0.y = b0p[k0 * X_LD + X_LD];
        v2f b1; b1.x = b1p[k0 * X_LD]; b1.y = b1p[k0 * X_LD + X_LD];
        acc0 = __builtin_amdgcn_wmma_f32_16x16x4_f32(false, a, false, b0, (short)0, acc0, false, false);
        acc1 = __builtin_amdgcn_wmma_f32_16x16x4_f32(false, a, false, b1, (short)0, acc1, false, false);
      }
    } else {                                           // atx = An^T @ x
      const float* ap = sAn + khalf * AN_LD + row;
      #pragma unroll
      for (int k0 = 0; k0 < NDIM; k0 += 4) {
        v2f a; a.x = ap[k0 * AN_LD]; a.y = ap[k0 * AN_LD + AN_LD];
        v2f b0; b0.x = b0p[k0 * X_LD]; b0.y = b0p[k0 * X_LD + X_LD];
        v2f b1; b1.x = b1p[k0 * X_LD]; b1.y = b1p[k0 * X_LD + X_LD];
        acc0 = __builtin_amdgcn_wmma_f32_16x16x4_f32(false, a, false, b0, (short)0, acc0, false, false);
        acc1 = __builtin_amdgcn_wmma_f32_16x16x4_f32(false, a, false, b1, (short)0, acc1, false, false);
      }
    }
    float* dst = (mat == 0) ? sax : satx;
    #pragma unroll
    for (int j = 0; j < 8; ++j) {
      const int m = tileM * 16 + ((lane < 16) ? j : (j + 8));
      dst[m * X_LD + col]      = acc0[j];
      dst[m * X_LD + 16 + col] = acc1[j];
    }
  }
  __syncthreads();

  // --- GEMM2: wave -> tileM = w>>1, tileN pair = (w&1)*2 + {0,1} ---
  {
    const int tileM = wave >> 1;
    const int npair = (wave & 1) * 2;
    const int row   = tileM * 16 + row16;
    const int nc0   = npair * 16 + col;
    const int nc1   = nc0 + 16;
    v8f acc0 = {}, acc1 = {};
    const float* srcs[3] = { sx, sax, satx };
    #pragma unroll
    for (int seg = 0; seg < 3; ++seg) {
      const float* ap  = srcs[seg] + row * X_LD + khalf;
      const float* bp0 = sW + (seg * CIN + khalf) * HDIM + nc0;
      const float* bp1 = bp0 + 16;
      #pragma unroll
      for (int k0 = 0; k0 < CIN; k0 += 4) {
        v2f a = *(const v2f*)(ap + k0);
        v2f b0; b0.x = bp0[k0 * HDIM]; b0.y = bp0[k0 * HDIM + HDIM];
        v2f b1; b1.x = bp1[k0 * HDIM]; b1.y = bp1[k0 * HDIM + HDIM];
        acc0 = __builtin_amdgcn_wmma_f32_16x16x4_f32(false, a, false, b0, (short)0, acc0, false, false);
        acc1 = __builtin_amdgcn_wmma_f32_16x16x4_f32(false, a, false, b1, (short)0, acc1, false, false);
      }
    }
    const float bias0 = sB[nc0], bias1 = sB[nc1];
    float* dst = xs + (size_t)blockIdx.x * NDIM * HDIM;
    #pragma unroll
    for (int j = 0; j < 8; ++j) {
      const int m = tileM * 16 + ((lane < 16) ? j : (j + 8));
      dst[m * HDIM + nc0] = acc0[j] + bias0;
      dst[m * HDIM + nc1] = acc1[j] + bias1;
    }
  }
}

// ---------------------------------------------------------------------------
// Kernel 3: per (b,n): multi-scale temporal conv (WMMA, dual-acc chains),
// relu, temporal LN, residual GEMM (WMMA, 4 indep accs), final LN, store.
// 512 threads = 16 waves; ~180KB LDS -> 1 workgroup per WGP (4 waves/SIMD).
// ---------------------------------------------------------------------------
__global__ void __launch_bounds__(512) stgcn_temporal(
    const float* __restrict__ xs, const float* __restrict__ x,
    const float* __restrict__ rw, const float* __restrict__ rb,
    const float* __restrict__ tng, const float* __restrict__ tnb,
    const float* __restrict__ lng, const float* __restrict__ lnb,
    const float* __restrict__ w1, const float* __restrict__ b1,
    const float* __restrict__ w3, const float* __restrict__ b3,
    const float* __restrict__ w5, const float* __restrict__ b5,
    const float* __restrict__ w7, const float* __restrict__ b7,
    float* __restrict__ out)
{
  extern __shared__ float smem[];
  float* sseq = smem;                            // (256+6)*68, pad 3 rows each side
  float* sout = sseq + (SDIM + 6) * SEQ_LD;      // 256*68
  float* sxin = sout + SDIM * SEQ_LD;            // 256*36
  float* smu  = sxin + SDIM * X_LD;              // 256
  float* sri  = smu + SDIM;                      // 256

  const int b    = blockIdx.x >> 7;              // / NDIM
  const int n    = blockIdx.x & (NDIM - 1);
  const int tid  = threadIdx.x;                  // 512 = 16 waves
  const int wave = tid >> 5;
  const int lane = tid & 31;
  const int row16 = lane & 15;
  const int khalf = (lane >> 4) * 2;
  const int col   = lane & 15;

  // --- stage xs[b,:,n,:] (row stride N*H, TDM-friendly 2D gather) and
  //     x[b,:,n,:] (row stride N*Cin) with hardware pitch padding ---
#if USE_TDM
  if (tid == 0) {
    tdm_load_2d((unsigned)(uintptr_t)(sseq + 3 * SEQ_LD),
                xs + ((size_t)b * SDIM * NDIM + n) * HDIM, HDIM, SDIM,
                (unsigned long long)NDIM * HDIM, /*pi=64dw*/5, /*pa=4dw*/3);
    tdm_load_2d((unsigned)(uintptr_t)sxin,
                x + ((size_t)b * SDIM * NDIM + n) * CIN, CIN, SDIM,
                (unsigned long long)NDIM * CIN, /*pi=32dw*/4, /*pa=4dw*/3);
  }
#else
  for (int i = tid; i < SDIM * HDIM / 4; i += 512) {   // 256 rows x 16 quads
    const int r = i >> 4, q = (i & 15) * 4;
    *(float4*)(sseq + (r + 3) * SEQ_LD + q) =
        *(const float4*)(xs + (((size_t)b * SDIM + r) * NDIM + n) * HDIM + q);
  }
  for (int i = tid; i < SDIM * CIN / 4; i += 512) {    // 256 rows x 8 quads
    const int r = i >> 3, q = (i & 7) * 4;
    *(float4*)(sxin + r * X_LD + q) =
        *(const float4*)(x + (((size_t)b * SDIM + r) * NDIM + n) * CIN + q);
  }
#endif
  if (tid < 6) {                                       // zero pad rows 0..2, 259..261
    const int pr = (tid < 3) ? tid : (tid + 256);
    for (int i = 0; i < HDIM; ++i) sseq[pr * SEQ_LD + i] = 0.f;
  }
#if USE_TDM
  __builtin_amdgcn_s_wait_tensorcnt(0);
#endif
  __syncthreads();

  // --- conv branches: wave -> tileT = wave (uniform), all 4 branches.
  //     collapsed (dt,h) K-loop with alternating dual accumulators ---
  {
    const int tileT = wave;
    const int trow  = tileT * 16 + row16;
    #pragma unroll
    for (int br = 0; br < 4; ++br) {
      const int ksz = 1 + 2 * br;
      const float* wp = (br == 0) ? w1 : (br == 1) ? w3 : (br == 2) ? w5 : w7;
      const float* bp = (br == 0) ? b1 : (br == 1) ? b3 : (br == 2) ? b5 : b7;
      const float* a0 = sseq + (trow - br + 3) * SEQ_LD + khalf;  // dt advances rows
      const float* bw = wp + khalf * HB + col;
      v8f accA = {}, accB = {};
      #pragma unroll
      for (int kk = 0; kk < 16 * 7; ++kk) {        // bounded; pruned by ksz check
        if (kk < 16 * ksz) {
          const int dt = kk >> 4;
          const int h0 = (kk & 15) * 4;
          v2f a = *(const v2f*)(a0 + dt * SEQ_LD + h0);
          v2f bb;
          bb.x = bw[(dt * HDIM + h0) * HB];
          bb.y = bw[(dt * HDIM + h0) * HB + HB];
          if (kk & 1)
            accB = __builtin_amdgcn_wmma_f32_16x16x4_f32(false, a, false, bb, (short)0, accB, false, false);
          else
            accA = __builtin_amdgcn_wmma_f32_16x16x4_f32(false, a, false, bb, (short)0, accA, false, false);
        }
      }
      v8f acc = accA + accB;
      const float bias = bp[col];
      #pragma unroll
      for (int j = 0; j < 8; ++j) {
        const int m = tileT * 16 + ((lane < 16) ? j : (j + 8));
        sout[m * SEQ_LD + br * HB + col] = fmaxf(acc[j] + bias, 0.f);
      }
    }
  }
  __syncthreads();

  // --- temporal layernorm, one row per thread (first 256 threads) ---
  if (tid < SDIM) {
    const int t = tid;
    float s = 0.f, sq = 0.f;
    for (int h = 0; h < HDIM; ++h) { float v = sout[t * SEQ_LD + h]; s += v; sq += v * v; }
    const float mu = s * (1.f / HDIM);
    const float ri = rsqrtf(sq * (1.f / HDIM) - mu * mu + EPSF);
    for (int h = 0; h < HDIM; ++h) {
      float v = sout[t * SEQ_LD + h];
      sout[t * SEQ_LD + h] = (v - mu) * ri * tng[h] + tnb[h];
    }
  }
  __syncthreads();

  // --- residual GEMM: wave -> tileT = wave, 4 independent tileN accs ---
  {
    const int tileT = wave;
    const int trow  = tileT * 16 + row16;
    const float* ap = sxin + trow * X_LD + khalf;
    v8f acc0 = {}, acc1 = {}, acc2 = {}, acc3 = {};
    const float* bw = rw + khalf * HDIM + col;
    #pragma unroll
    for (int k0 = 0; k0 < CIN; k0 += 4) {
      v2f a = *(const v2f*)(ap + k0);
      v2f bb0; bb0.x = bw[k0 * HDIM];      bb0.y = bw[k0 * HDIM + HDIM];
      v2f bb1; bb1.x = bw[k0 * HDIM + 16]; bb1.y = bw[k0 * HDIM + HDIM + 16];
      v2f bb2; bb2.x = bw[k0 * HDIM + 32]; bb2.y = bw[k0 * HDIM + HDIM + 32];
      v2f bb3; bb3.x = bw[k0 * HDIM + 48]; bb3.y = bw[k0 * HDIM + HDIM + 48];
      acc0 = __builtin_amdgcn_wmma_f32_16x16x4_f32(false, a, false, bb0, (short)0, acc0, false, false);
      acc1 = __builtin_amdgcn_wmma_f32_16x16x4_f32(false, a, false, bb1, (short)0, acc1, false, false);
      acc2 = __builtin_amdgcn_wmma_f32_16x16x4_f32(false, a, false, bb2, (short)0, acc2, false, false);
      acc3 = __builtin_amdgcn_wmma_f32_16x16x4_f32(false, a, false, bb3, (short)0, acc3, false, false);
    }
    #pragma unroll
    for (int j = 0; j < 8; ++j) {
      const int m = tileT * 16 + ((lane < 16) ? j : (j + 8));
      sout[m * SEQ_LD + col]      += acc0[j] + rb[col];
      sout[m * SEQ_LD + col + 16] += acc1[j] + rb[col + 16];
      sout[m * SEQ_LD + col + 32] += acc2[j] + rb[col + 32];
      sout[m * SEQ_LD + col + 48] += acc3[j] + rb[col + 48];
    }
  }
  __syncthreads();

  // --- final LN stats per row ---
  if (tid < SDIM) {
    const int t = tid;
    float s = 0.f, sq = 0.f;
    for (int h = 0; h < HDIM; ++h) { float v = sout[t * SEQ_LD + h]; s += v; sq += v * v; }
    const float mu = s * (1.f / HDIM);
    smu[t] = mu;
    sri[t] = rsqrtf(sq * (1.f / HDIM) - mu * mu + EPSF);
  }
  __syncthreads();

  // --- normalize + coalesced float4 store ---
  for (int idx = tid; idx < SDIM * HDIM / 4; idx += 512) {
    const int t  = idx >> 4;                   // 16 float4 per row
    const int c4 = (idx & 15) * 4;
    const float mu = smu[t], ri = sri[t];
    float4 o;
    o.x = (sout[t * SEQ_LD + c4 + 0] - mu) * ri * lng[c4 + 0] + lnb[c4 + 0];
    o.y = (sout[t * SEQ_LD + c4 + 1] - mu) * ri * lng[c4 + 1] + lnb[c4 + 1];
    o.z = (sout[t * SEQ_LD + c4 + 2] - mu) * ri * lng[c4 + 2] + lnb[c4 + 2];
    o.w = (sout[t * SEQ_LD + c4 + 3] - mu) * ri * lng[c4 + 3] + lnb[c4 + 3];
    *(float4*)(out + (((size_t)b * SDIM + t) * NDIM + n) * HDIM + c4) = o;
  }
}

// ---------------------------------------------------------------------------
extern "C" void kernel_launch(void* const* d_in, const int* in_sizes, int n_in,
                              void* d_out, int out_size, void* d_ws, size_t ws_size,
                              hipStream_t stream) {
  const float* x   = (const float*)d_in[0];
  const float* adj = (const float*)d_in[1];
  const float* thw = (const float*)d_in[2];
  const float* thb = (const float*)d_in[3];
  const float* phw = (const float*)d_in[4];
  const float* phb = (const float*)d_in[5];
  const float* psw = (const float*)d_in[6];
  const float* psb = (const float*)d_in[7];
  const float* rw  = (const float*)d_in[8];
  const float* rb  = (const float*)d_in[9];
  const float* tng = (const float*)d_in[10];
  const float* tnb = (const float*)d_in[11];
  const float* lng = (const float*)d_in[12];
  const float* lnb = (const float*)d_in[13];
  const float* w1  = (const float*)d_in[14];
  const float* b1  = (const float*)d_in[15];
  const float* w3  = (const float*)d_in[16];
  const float* b3  = (const float*)d_in[17];
  const float* w5  = (const float*)d_in[18];
  const float* b5  = (const float*)d_in[19];
  const float* w7  = (const float*)d_in[20];
  const float* b7  = (const float*)d_in[21];
  float* out = (float*)d_out;

  float* xs = (float*)d_ws;                               // (B,S,N,H) f32
  float* An = xs + (size_t)BDIM * SDIM * NDIM * HDIM;     // 128*128 f32

  stgcn_adj_norm<<<1, NDIM, 0, stream>>>(adj, An);

  const size_t smemB = (size_t)(NDIM * AN_LD + 3 * NDIM * X_LD +
                                3 * CIN * HDIM + HDIM) * sizeof(float);
  stgcn_spatial<<<BDIM * SDIM, 512, smemB, stream>>>(
      x, An, thw, thb, phw, phb, psw, psb, xs);

  const size_t smemC = (size_t)((SDIM + 6) * SEQ_LD + SDIM * SEQ_LD +
                                SDIM * X_LD + 2 * SDIM) * sizeof(float);
  stgcn_temporal<<<BDIM * NDIM, 512, smemC, stream>>>(
      xs, x, rw, rb, tng, tnb, lng, lnb, w1, b1, w3, b3, w5, b5, w7, b7, out);
}